// GraphConvolution_35476429865958
// MI455X (gfx1250) — compile-verified
//
#include <hip/hip_runtime.h>

// Graph convolution, fused: out = X@W1 + b1 + (sum_j X[nbr_j])@W2 + b2
// fp32 end-to-end via V_WMMA_F32_16X16X4_F32.
// M=32 nodes/block so each B fragment feeds 2 WMMAs; W pre-packed in ws into
// k-pair-interleaved layout so each B fragment is one coalesced global_load_b64.

#define NNODES 100000
#define DEG    16
#define DIN    192
#define DOUT   192
#define AS     196   // padded LDS row stride in floats (196 % 64 == 4 -> conflict-free)
#define MB     32    // nodes per block

typedef __attribute__((ext_vector_type(2))) float v2f;
typedef __attribute__((ext_vector_type(8))) float v8f;

// Pack W (row-major [k][n]) into Wp: for k-pair p, Wp[p*2*DOUT + 2n + {0,1}] =
// W[2p][n], W[2p+1][n].  blockIdx.y selects W1 (->Wp) or W2 (->Wp+DIN*DOUT).
__global__ __launch_bounds__(192)
void pack_w_kernel(const float* __restrict__ W1, const float* __restrict__ W2,
                   float* __restrict__ Wp) {
    const int p = blockIdx.x;      // 0..95   k-pair
    const int n = threadIdx.x;     // 0..191  column
    const float* W = (blockIdx.y == 0) ? W1 : W2;
    float* dst = Wp + (size_t)blockIdx.y * DIN * DOUT + (size_t)p * 2 * DOUT + 2 * n;
    dst[0] = W[(size_t)(2 * p)     * DOUT + n];
    dst[1] = W[(size_t)(2 * p + 1) * DOUT + n];
}

template <bool PACKED>
__global__ __launch_bounds__(128)
void GraphConvolution_35476429865958_kernel(
    const int*   __restrict__ nbr,   // [N, DEG]
    const float* __restrict__ X,     // [N, DIN]
    const float* __restrict__ W1,    // [DIN, DOUT]   (used when !PACKED)
    const float* __restrict__ B1,    // [DOUT]
    const float* __restrict__ W2,    // [DIN, DOUT]   (used when !PACKED)
    const float* __restrict__ B2,    // [DOUT]
    const float* __restrict__ Wp,    // packed W1|W2  (used when PACKED)
    float*       __restrict__ out)   // [N, DOUT]
{
    __shared__ float A1[MB * AS];    // own-node rows
    __shared__ float A2[MB * AS];    // neighbor-sum rows

    const int tid   = threadIdx.x;
    const int node0 = blockIdx.x * MB;

    // ---------------- Stage 1: load own rows + gather/sum neighbours ----------------
    {
        const int row  = tid >> 2;     // 0..31  (node within block)
        const int part = tid & 3;      // 0..3   (48 floats each)
        const int cb   = part * 48;

        const float* src = X + (size_t)(node0 + row) * DIN + cb;
        float* d1 = &A1[row * AS + cb];
        #pragma unroll
        for (int i = 0; i < 12; ++i)
            *(float4*)&d1[i * 4] = *(const float4*)&src[i * 4];

        float acc[48];
        #pragma unroll
        for (int i = 0; i < 48; ++i) acc[i] = 0.f;

        const int* nrow = nbr + (size_t)(node0 + row) * DEG;
        for (int j = 0; j < DEG; ++j) {
            const float* s = X + (size_t)nrow[j] * DIN + cb;
            #pragma unroll
            for (int i = 0; i < 12; ++i) {
                float4 v = *(const float4*)&s[i * 4];
                acc[i * 4 + 0] += v.x;
                acc[i * 4 + 1] += v.y;
                acc[i * 4 + 2] += v.z;
                acc[i * 4 + 3] += v.w;
            }
        }
        float* d2 = &A2[row * AS + cb];
        #pragma unroll
        for (int i = 0; i < 12; ++i) {
            float4 v = { acc[i*4+0], acc[i*4+1], acc[i*4+2], acc[i*4+3] };
            *(float4*)&d2[i * 4] = v;
        }
    }
    __syncthreads();

    // ---------------- Stage 2: WMMA f32 16x16x4 ----------------
    // Wave w owns output columns [w*48, w*48+48) = 3 N-tiles, for both M-tiles.
    const int wave  = tid >> 5;
    const int lane  = tid & 31;
    const int laneN = lane & 15;     // A: row M ; B/C: col N
    const int laneH = lane >> 4;     // K sub-pair select (A/B), M-half select (C)

    const int col0 = wave * 48 + laneN;
    v8f c[2][3];
    #pragma unroll
    for (int m = 0; m < 2; ++m)
        #pragma unroll
        for (int t = 0; t < 3; ++t) c[m][t] = (v8f){};

    const float* a10 = &A1[laneN * AS + laneH * 2];
    const float* a11 = &A1[(16 + laneN) * AS + laneH * 2];
    const float* a20 = &A2[laneN * AS + laneH * 2];
    const float* a21 = &A2[(16 + laneN) * AS + laneH * 2];

    const float* p1 = Wp + 2 * col0;                          // packed W1
    const float* p2 = Wp + (size_t)DIN * DOUT + 2 * col0;     // packed W2
    const float* w1p = W1 + col0;                             // fallback
    const float* w2p = W2 + col0;

    // Phase 1: += Xtile @ W1
    #pragma unroll 2
    for (int kk = 0; kk < DIN; kk += 4) {
        v2f b0, b1v, b2v;
        if (PACKED) {
            const float* w = p1 + (size_t)((kk >> 1) + laneH) * (2 * DOUT);
            b0  = *(const v2f*)(w);         // cols col0    : lanes 0-15 contiguous 128B
            b1v = *(const v2f*)(w + 32);    // cols col0+16
            b2v = *(const v2f*)(w + 64);    // cols col0+32
        } else {
            const float* w = w1p + (size_t)(kk + laneH * 2) * DOUT;
            b0  = (v2f){ w[0],  w[DOUT]      };
            b1v = (v2f){ w[16], w[DOUT + 16] };
            b2v = (v2f){ w[32], w[DOUT + 32] };
        }
        v2f a0 = *(const v2f*)&a10[kk];
        v2f a1 = *(const v2f*)&a11[kk];
        c[0][0] = __builtin_amdgcn_wmma_f32_16x16x4_f32(false, a0, false, b0,  (short)0, c[0][0], false, false);
        c[1][0] = __builtin_amdgcn_wmma_f32_16x16x4_f32(false, a1, false, b0,  (short)0, c[1][0], false, false);
        c[0][1] = __builtin_amdgcn_wmma_f32_16x16x4_f32(false, a0, false, b1v, (short)0, c[0][1], false, false);
        c[1][1] = __builtin_amdgcn_wmma_f32_16x16x4_f32(false, a1, false, b1v, (short)0, c[1][1], false, false);
        c[0][2] = __builtin_amdgcn_wmma_f32_16x16x4_f32(false, a0, false, b2v, (short)0, c[0][2], false, false);
        c[1][2] = __builtin_amdgcn_wmma_f32_16x16x4_f32(false, a1, false, b2v, (short)0, c[1][2], false, false);
    }
    // Phase 2: += aggr @ W2 (same accumulators)
    #pragma unroll 2
    for (int kk = 0; kk < DIN; kk += 4) {
        v2f b0, b1v, b2v;
        if (PACKED) {
            const float* w = p2 + (size_t)((kk >> 1) + laneH) * (2 * DOUT);
            b0  = *(const v2f*)(w);
            b1v = *(const v2f*)(w + 32);
            b2v = *(const v2f*)(w + 64);
        } else {
            const float* w = w2p + (size_t)(kk + laneH * 2) * DOUT;
            b0  = (v2f){ w[0],  w[DOUT]      };
            b1v = (v2f){ w[16], w[DOUT + 16] };
            b2v = (v2f){ w[32], w[DOUT + 32] };
        }
        v2f a0 = *(const v2f*)&a20[kk];
        v2f a1 = *(const v2f*)&a21[kk];
        c[0][0] = __builtin_amdgcn_wmma_f32_16x16x4_f32(false, a0, false, b0,  (short)0, c[0][0], false, false);
        c[1][0] = __builtin_amdgcn_wmma_f32_16x16x4_f32(false, a1, false, b0,  (short)0, c[1][0], false, false);
        c[0][1] = __builtin_amdgcn_wmma_f32_16x16x4_f32(false, a0, false, b1v, (short)0, c[0][1], false, false);
        c[1][1] = __builtin_amdgcn_wmma_f32_16x16x4_f32(false, a1, false, b1v, (short)0, c[1][1], false, false);
        c[0][2] = __builtin_amdgcn_wmma_f32_16x16x4_f32(false, a0, false, b2v, (short)0, c[0][2], false, false);
        c[1][2] = __builtin_amdgcn_wmma_f32_16x16x4_f32(false, a1, false, b2v, (short)0, c[1][2], false, false);
    }

    // ---------------- Epilogue: bias + store ----------------
    // C layout: VGPR r -> (M=r, N=lane) lanes 0-15, (M=8+r, N=lane-16) lanes 16-31.
    const int mbase = laneH * 8;
    #pragma unroll
    for (int m = 0; m < 2; ++m) {
        #pragma unroll
        for (int t = 0; t < 3; ++t) {
            const int col  = col0 + t * 16;
            const float bias = B1[col] + B2[col];
            #pragma unroll
            for (int r = 0; r < 8; ++r) {
                out[(size_t)(node0 + m * 16 + mbase + r) * DOUT + col] = c[m][t][r] + bias;
            }
        }
    }
}

extern "C" void kernel_launch(void* const* d_in, const int* in_sizes, int n_in,
                              void* d_out, int out_size, void* d_ws, size_t ws_size,
                              hipStream_t stream) {
    const int*   nbr = (const int*)  d_in[0];
    const float* X   = (const float*)d_in[1];
    const float* W1  = (const float*)d_in[2];
    const float* b1  = (const float*)d_in[3];
    const float* W2  = (const float*)d_in[4];
    const float* b2  = (const float*)d_in[5];
    float* out = (float*)d_out;

    const size_t pack_bytes = (size_t)2 * DIN * DOUT * sizeof(float);  // 294,912 B
    dim3 grid(NNODES / MB);   // 3125 blocks of 32 nodes (exact)
    dim3 block(128);          // 4 wave32s

    if (ws_size >= pack_bytes) {
        float* Wp = (float*)d_ws;
        pack_w_kernel<<<dim3(DIN / 2, 2), 192, 0, stream>>>(W1, W2, Wp);
        GraphConvolution_35476429865958_kernel<true><<<grid, block, 0, stream>>>(
            nbr, X, W1, b1, W2, b2, Wp, out);
    } else {
        GraphConvolution_35476429865958_kernel<false><<<grid, block, 0, stream>>>(
            nbr, X, W1, b1, W2, b2, (const float*)nullptr, out);
    }
}